// MovingHandwritingVAE_72224170050132
// MI455X (gfx1250) — compile-verified
//
#include <hip/hip_runtime.h>
#include <cstdint>

// ---------------------------------------------------------------------------
// Reference collapses to: per batch b, recover idx from the one-hot row, then
// out[b, pi, pj] = image[b, idx/68 + pi - 14, idx%68 + pj - 14] (0 if OOB).
// Pure gather, ~1.6 MB total traffic -> HBM roofline ~70ns; launch-bound.
// No matrix op survives the collapse, so no WMMA (im2col-WMMA would add
// ~580 MB of traffic). CDNA5-specific path: async global->LDS staging
// (global_load_async_to_lds_b128 / ASYNCcnt) of the one-hot row.
// ---------------------------------------------------------------------------

constexpr int FULL   = 68;
constexpr int NFULL  = FULL * FULL;   // 4624
constexpr int SLEN   = 28;
constexpr int NPATCH = SLEN * SLEN;   // 784
constexpr int PAD    = SLEN / 2;      // 14
constexpr int BATCH  = 64;
constexpr int NTHR   = 256;           // 8 wave32s
constexpr int NCHUNK = NFULL / 4;     // 1156 x 16-byte async transfers

__global__ __launch_bounds__(NTHR)
void patch_gather_kernel(const float* __restrict__ one_hot,
                         const float* __restrict__ image,
                         float* __restrict__ out) {
  __shared__ __align__(16) float s_oh[NFULL];
  __shared__ int s_idx;

  const int b   = blockIdx.x;
  const int tid = threadIdx.x;

  if (tid == 0) s_idx = 0;  // defensive init (exactly one hot element expected)

  // ---- Phase 0: async-copy one_hot row b into LDS (gfx1250 async path). ----
  // 4624 floats = 1156 b128 chunks; each active lane copies 16B global->LDS.
  const float4* row4 = (const float4*)(one_hot + (size_t)b * NFULL);
  for (int c = tid; c < NCHUNK; c += NTHR) {
    // Low 32 bits of a generic LDS pointer are the LDS byte offset (ISA 10.2).
    unsigned lds_addr = (unsigned)(uint64_t)(const void*)&s_oh[c * 4];
    unsigned long long gaddr = (unsigned long long)(const void*)(row4 + c);
    asm volatile("global_load_async_to_lds_b128 %0, %1, off"
                 :: "v"(lds_addr), "v"(gaddr)
                 : "memory");
  }
  asm volatile("s_wait_asynccnt 0" ::: "memory");  // this wave's copies done
  __syncthreads();                                  // all waves' copies visible

  // ---- Phase 1: find the hot index (exactly one element is 1.0f). ----
  // Vectorized LDS scan: ds_load_b128, ~5 iterations/thread instead of ~18.
  const float4* s4 = (const float4*)s_oh;
  for (int q = tid; q < NCHUNK; q += NTHR) {
    const float4 v = s4[q];
    if (v.x > 0.5f) s_idx = 4 * q + 0;
    if (v.y > 0.5f) s_idx = 4 * q + 1;
    if (v.z > 0.5f) s_idx = 4 * q + 2;
    if (v.w > 0.5f) s_idx = 4 * q + 3;
  }
  __syncthreads();

  const int idx = s_idx;
  const int y0  = idx / FULL - PAD;
  const int x0  = idx % FULL - PAD;

  // ---- Phase 2: bounds-checked 28x28 patch gather (coalesced rows). ----
  const float* img = image + (size_t)b * NFULL;
  float*       dst = out   + (size_t)b * NPATCH;

  for (int p = tid; p < NPATCH; p += NTHR) {
    const int pi = p / SLEN;
    const int pj = p % SLEN;           // consecutive lanes -> consecutive cols
    const int r  = y0 + pi;
    const int c  = x0 + pj;
    float v = 0.0f;
    if ((unsigned)r < (unsigned)FULL && (unsigned)c < (unsigned)FULL)
      v = img[r * FULL + c];
    dst[p] = v;
  }
}

extern "C" void kernel_launch(void* const* d_in, const int* in_sizes, int n_in,
                              void* d_out, int out_size, void* d_ws, size_t ws_size,
                              hipStream_t stream) {
  const float* one_hot = (const float*)d_in[0];  // (64, 4624) f32
  const float* image   = (const float*)d_in[1];  // (64, 1, 68, 68) f32
  // d_in[2] (w_id) is the identity im2col filter by construction — unused:
  // exploiting that identity is precisely the optimization.
  float* out = (float*)d_out;                    // (64, 1, 28, 28) f32

  patch_gather_kernel<<<BATCH, NTHR, 0, stream>>>(one_hot, image, out);
}